// PAConv_32263794328117
// MI455X (gfx1250) — compile-verified
//
#include <hip/hip_runtime.h>
#include <hip/hip_bf16.h>

static constexpr int B_SZ  = 8;
static constexpr int N_PTS = 1024;
static constexpr int K_NB  = 20;
static constexpr int M_K   = 8;
static constexpr int BN    = B_SZ * N_PTS;          // 8192
static constexpr int BNK   = BN * K_NB;             // 163840
static constexpr float EPSV = 1e-5f;
static constexpr float NEG_INF = -3.4e38f;

typedef float v2f __attribute__((ext_vector_type(2)));
typedef float v8f __attribute__((ext_vector_type(8)));

// ---------------- workspace layout (float offsets) ----------------
static constexpr size_t OFF_PTS   = 0;                                  // BN*3
static constexpr size_t OFF_SQ    = OFF_PTS   + (size_t)BN * 3;         // BN
static constexpr size_t OFF_IDX   = OFF_SQ    + (size_t)BN;             // BNK (ints)
static constexpr size_t OFF_XYZ   = OFF_IDX   + (size_t)BNK;            // BNK*6
static constexpr size_t OFF_F64   = OFF_XYZ   + (size_t)BNK * 6;        // BN*64
static constexpr size_t OFF_F128  = OFF_F64   + (size_t)BN * 64;        // BN*128
static constexpr size_t OFF_BIG   = OFF_F128  + (size_t)BN * 128;       // BN*1024
static constexpr size_t OFF_H     = OFF_BIG   + (size_t)BN * 1024;      // BNK*16
static constexpr size_t OFF_SCORE = OFF_H     + (size_t)BNK * 16;       // BNK*8
static constexpr size_t OFF_FMAX  = OFF_SCORE + (size_t)BNK * 8;        // 16*1024 (rows 8..15 zero-padded)
static constexpr size_t OFF_F512  = OFF_FMAX  + (size_t)16 * 1024;      // 16*512
static constexpr size_t OFF_STATS = OFF_F512  + (size_t)16 * 512;       // 2048

// ---------------- kernels ----------------

__global__ void k_build_pts(const float* __restrict__ x, float* __restrict__ pts,
                            float* __restrict__ sq) {
    int i = blockIdx.x * blockDim.x + threadIdx.x;   // b*N + n
    if (i >= BN) return;
    int b = i >> 10, n = i & (N_PTS - 1);
    const float* xb = x + (size_t)b * 3 * N_PTS;
    float a = xb[n], c = xb[N_PTS + n], d = xb[2 * N_PTS + n];
    pts[(size_t)i * 3 + 0] = a;
    pts[(size_t)i * 3 + 1] = c;
    pts[(size_t)i * 3 + 2] = d;
    sq[i] = a * a + c * c + d * d;
}

#define KNN_T 128
__global__ void k_knn(const float* __restrict__ pts, const float* __restrict__ sq,
                      int* __restrict__ idx) {
    __shared__ float d[N_PTS];
    __shared__ float sv[KNN_T];
    __shared__ int   si[KNN_T];
    int bn = blockIdx.x;
    int b = bn >> 10, n = bn & (N_PTS - 1);
    const float* pb = pts + (size_t)b * N_PTS * 3;
    float px = pb[n * 3 + 0], py = pb[n * 3 + 1], pz = pb[n * 3 + 2];
    float sn = sq[bn];
    for (int m = threadIdx.x; m < N_PTS; m += KNN_T) {
        float dot = px * pb[m * 3 + 0] + py * pb[m * 3 + 1] + pz * pb[m * 3 + 2];
        d[m] = 2.f * dot - sn - sq[b * N_PTS + m];     // = -||pn - pm||^2
    }
    __syncthreads();
    for (int sel = 0; sel < K_NB; ++sel) {
        float bv = NEG_INF; int bi = 0x7fffffff;
        for (int m = threadIdx.x; m < N_PTS; m += KNN_T) {
            float v = d[m];
            if (v > bv || (v == bv && m < bi)) { bv = v; bi = m; }
        }
        sv[threadIdx.x] = bv; si[threadIdx.x] = bi;
        __syncthreads();
        for (int s = KNN_T / 2; s > 0; s >>= 1) {
            if (threadIdx.x < s) {
                float ov = sv[threadIdx.x + s]; int oi = si[threadIdx.x + s];
                if (ov > sv[threadIdx.x] ||
                    (ov == sv[threadIdx.x] && oi < si[threadIdx.x])) {
                    sv[threadIdx.x] = ov; si[threadIdx.x] = oi;
                }
            }
            __syncthreads();
        }
        if (threadIdx.x == 0) {
            idx[(size_t)bn * K_NB + sel] = si[0];
            d[si[0]] = NEG_INF;
        }
        __syncthreads();
    }
}

__global__ void k_xyz(const float* __restrict__ pts, const int* __restrict__ idx,
                      float* __restrict__ xyz) {
    int t = blockIdx.x * blockDim.x + threadIdx.x;   // bn*K + k
    if (t >= BNK) return;
    int bn = t / K_NB;
    int j = idx[t];
    int b = bn >> 10;
    const float* c0 = pts + (size_t)bn * 3;
    const float* nb = pts + (size_t)(b * N_PTS + j) * 3;
    float* o = xyz + (size_t)t * 6;
    o[0] = nb[0] - c0[0]; o[1] = nb[1] - c0[1]; o[2] = nb[2] - c0[2];
    o[3] = nb[0];         o[4] = nb[1];         o[5] = nb[2];
}

// out[r,c] = sum_d in[r,d] * w[c,d]   (w row-major (Cout, Cin))
__global__ void k_linmap(const float* __restrict__ in, const float* __restrict__ w,
                         float* __restrict__ out, int rows, int Cin, int Cout) {
    size_t i = (size_t)blockIdx.x * blockDim.x + threadIdx.x;
    if (i >= (size_t)rows * Cout) return;
    int c = (int)(i % Cout);
    size_t r = i / Cout;
    const float* ip = in + r * Cin;
    const float* wp = w + (size_t)c * Cin;
    float a = 0.f;
    for (int d = 0; d < Cin; ++d) a += ip[d] * wp[d];
    out[i] = a;
}

__global__ void k_zero(float* __restrict__ p, int n) {
    int i = blockIdx.x * blockDim.x + threadIdx.x;
    if (i < n) p[i] = 0.f;
}

// channel sums + sums of squares (C must be a power of two)
__global__ void k_stats(const float* __restrict__ y, int rows, int C,
                        float* __restrict__ sums) {
    __shared__ float ls[2048];
    for (int i = threadIdx.x; i < 2 * C; i += blockDim.x) ls[i] = 0.f;
    __syncthreads();
    size_t total = (size_t)rows * C;
    int cm = C - 1;
    for (size_t i = (size_t)blockIdx.x * blockDim.x + threadIdx.x; i < total;
         i += (size_t)gridDim.x * blockDim.x) {
        float v = y[i];
        int c = (int)i & cm;
        atomicAdd(&ls[c], v);
        atomicAdd(&ls[C + c], v * v);
    }
    __syncthreads();
    for (int i = threadIdx.x; i < 2 * C; i += blockDim.x) atomicAdd(&sums[i], ls[i]);
}

__global__ void k_bn_apply(float* __restrict__ y, int rows, int C,
                           const float* __restrict__ sums, const float* __restrict__ g,
                           const float* __restrict__ bb, int relu) {
    size_t total = (size_t)rows * C;
    size_t i = (size_t)blockIdx.x * blockDim.x + threadIdx.x;
    if (i >= total) return;
    int c = (int)i & (C - 1);
    float cnt = (float)rows;
    float mean = sums[c] / cnt;
    float var = sums[C + c] / cnt - mean * mean;     // population variance (jnp.var)
    float o = g[c] * (y[i] - mean) * rsqrtf(var + EPSV) + bb[c];
    if (relu) o = fmaxf(o, 0.f);
    y[i] = o;
}

// Out(R x Cout) = A(R x Cin) @ W
// Requirements: R % 16 == 0, Cout % 32 == 0, CIN % 4 == 0 (no predication anywhere).
// wT=0: W row-major (Cin,Cout); wT=1: W row-major (Cout,Cin).
// One 16x32 output tile per wave32 (two accumulators share the A fragment).
template <int CIN>
__global__ void k_gemm_wmma(const float* __restrict__ A, const float* __restrict__ W,
                            float* __restrict__ Out, int R, int Cout, int wT) {
    int wave = threadIdx.x >> 5;
    int lane = threadIdx.x & 31;
    int tilesN = Cout >> 5;
    int t = blockIdx.x * (blockDim.x >> 5) + wave;
    if (t >= (R >> 4) * tilesN) return;          // whole-wave exit: EXEC stays all-ones
    int tm = t / tilesN, tn = t % tilesN;
    int row0 = tm << 4, col0 = tn << 5;
    int half = lane >> 4, l = lane & 15;         // ISA 16x16x4 f32 fragment layout
    const float* Arow = A + (size_t)(row0 + l) * CIN + half * 2;
    v8f acc0 = {}, acc1 = {};
    if (wT) {
        const float* W0 = W + (size_t)(col0 + l) * CIN + half * 2;
        const float* W1 = W0 + (size_t)16 * CIN;
#pragma unroll 8
        for (int kk = 0; kk < CIN / 4; ++kk) {
            v2f a  = *(const v2f*)(Arow + kk * 4);
            v2f b0 = *(const v2f*)(W0 + kk * 4);
            v2f b1 = *(const v2f*)(W1 + kk * 4);
            acc0 = __builtin_amdgcn_wmma_f32_16x16x4_f32(false, a, false, b0,
                                                         (short)0, acc0, false, false);
            acc1 = __builtin_amdgcn_wmma_f32_16x16x4_f32(false, a, false, b1,
                                                         (short)0, acc1, false, false);
        }
    } else {
        const float* Wc = W + (size_t)(half * 2) * Cout + col0 + l;
#pragma unroll 8
        for (int kk = 0; kk < CIN / 4; ++kk) {
            v2f a = *(const v2f*)(Arow + kk * 4);
            const float* wp = Wc + (size_t)(kk * 4) * Cout;
            v2f b0, b1;
            b0.x = wp[0];            b0.y = wp[(size_t)Cout];
            b1.x = wp[16];           b1.y = wp[(size_t)Cout + 16];
            acc0 = __builtin_amdgcn_wmma_f32_16x16x4_f32(false, a, false, b0,
                                                         (short)0, acc0, false, false);
            acc1 = __builtin_amdgcn_wmma_f32_16x16x4_f32(false, a, false, b1,
                                                         (short)0, acc1, false, false);
        }
    }
    float* Orow = Out + (size_t)(row0 + half * 8) * Cout + col0 + l;
#pragma unroll 8
    for (int r8 = 0; r8 < 8; ++r8) {             // VGPR r8: M = r8 (+8 for lanes>=16)
        Orow[(size_t)r8 * Cout]      = acc0[r8];
        Orow[(size_t)r8 * Cout + 16] = acc1[r8];
    }
}

// s = softmax(h(16) @ w2^T(8,16) + b2) per row
__global__ void k_score2(const float* __restrict__ h, const float* __restrict__ w2,
                         const float* __restrict__ b2, float* __restrict__ score,
                         int rows) {
    int r = blockIdx.x * blockDim.x + threadIdx.x;
    if (r >= rows) return;
    float hv[16];
#pragma unroll
    for (int c = 0; c < 16; ++c) hv[c] = h[(size_t)r * 16 + c];
    float s[M_K]; float mx = NEG_INF;
#pragma unroll
    for (int m = 0; m < M_K; ++m) {
        float a = b2[m];
#pragma unroll
        for (int c = 0; c < 16; ++c) a += hv[c] * w2[m * 16 + c];
        s[m] = a; mx = fmaxf(mx, a);
    }
    float sum = 0.f;
#pragma unroll
    for (int m = 0; m < M_K; ++m) { s[m] = __expf(s[m] - mx); sum += s[m]; }
    float inv = 1.f / sum;
#pragma unroll
    for (int m = 0; m < M_K; ++m) score[(size_t)r * M_K + m] = s[m] * inv;
}

// out[b,n,o] = sum_{k,m} score[b,n,k,m] * fM[b, idx[b,n,k], m, o]  (blockDim == Cf)
__global__ void k_assemble(const float* __restrict__ score, const float* __restrict__ fM,
                           const int* __restrict__ idx, float* __restrict__ outp,
                           int Cf) {
    __shared__ float ss[K_NB * M_K];
    __shared__ int   sj[K_NB];
    int bn = blockIdx.x;
    int b = bn >> 10;
    for (int i = threadIdx.x; i < K_NB * M_K; i += blockDim.x)
        ss[i] = score[(size_t)bn * K_NB * M_K + i];
    for (int i = threadIdx.x; i < K_NB; i += blockDim.x)
        sj[i] = idx[(size_t)bn * K_NB + i];
    __syncthreads();
    int o = threadIdx.x;
    float acc = 0.f;
    for (int k = 0; k < K_NB; ++k) {
        const float* fp = fM + (size_t)(b * N_PTS + sj[k]) * M_K * Cf;
        const float* sc = ss + k * M_K;
#pragma unroll
        for (int m = 0; m < M_K; ++m) acc += sc[m] * fp[m * Cf + o];
    }
    outp[(size_t)bn * Cf + o] = acc;
}

__global__ void k_colmax(const float* __restrict__ big, float* __restrict__ fm) {
    int i = blockIdx.x * blockDim.x + threadIdx.x;   // b*1024 + c
    if (i >= B_SZ * 1024) return;
    int b = i >> 10, c = i & 1023;
    const float* p = big + (size_t)b * N_PTS * 1024 + c;
    float m = NEG_INF;
    for (int n = 0; n < N_PTS; ++n) m = fmaxf(m, p[(size_t)n * 1024]);
    fm[i] = m;
}

__global__ void k_lin2(const float* __restrict__ f, const float* __restrict__ w,
                       const float* __restrict__ bb, float* __restrict__ out) {
    int i = blockIdx.x * blockDim.x + threadIdx.x;   // b*40 + o
    if (i >= B_SZ * 40) return;
    int b = i / 40, o = i % 40;
    float a = bb[o];
    const float* fp = f + b * 512;
    const float* wp = w + o * 512;
    for (int c = 0; c < 512; ++c) a += fp[c] * wp[c];
    out[i] = a;
}

// ---------------- host orchestration ----------------

extern "C" void kernel_launch(void* const* d_in, const int* in_sizes, int n_in,
                              void* d_out, int out_size, void* d_ws, size_t ws_size,
                              hipStream_t stream) {
    const float* x       = (const float*)d_in[0];
    const float* conv1_w = (const float*)d_in[1];
    const float* bn1_g   = (const float*)d_in[2];
    const float* bn1_b   = (const float*)d_in[3];
    const float* mat2    = (const float*)d_in[4];
    const float* mat3    = (const float*)d_in[5];
    const float* mat4    = (const float*)d_in[6];
    const float* sn_w1[3] = {(const float*)d_in[7],  (const float*)d_in[12], (const float*)d_in[17]};
    const float* sn_g [3] = {(const float*)d_in[8],  (const float*)d_in[13], (const float*)d_in[18]};
    const float* sn_b [3] = {(const float*)d_in[9],  (const float*)d_in[14], (const float*)d_in[19]};
    const float* sn_w2[3] = {(const float*)d_in[10], (const float*)d_in[15], (const float*)d_in[20]};
    const float* sn_b2[3] = {(const float*)d_in[11], (const float*)d_in[16], (const float*)d_in[21]};
    const float* bn2_g = (const float*)d_in[22];
    const float* bn2_b = (const float*)d_in[23];
    const float* bn3_g = (const float*)d_in[24];
    const float* bn3_b = (const float*)d_in[25];
    const float* bn4_g = (const float*)d_in[26];
    const float* bn4_b = (const float*)d_in[27];
    const float* conv5_w = (const float*)d_in[28];
    const float* bn5_g = (const float*)d_in[29];
    const float* bn5_b = (const float*)d_in[30];
    const float* lin1_w = (const float*)d_in[31];
    const float* bn6_g = (const float*)d_in[32];
    const float* bn6_b = (const float*)d_in[33];
    const float* lin2_w = (const float*)d_in[34];
    const float* lin2_b = (const float*)d_in[35];

    float* ws    = (float*)d_ws;
    float* pts   = ws + OFF_PTS;
    float* sq    = ws + OFF_SQ;
    int*   idx   = (int*)(ws + OFF_IDX);
    float* xyz   = ws + OFF_XYZ;
    float* f64   = ws + OFF_F64;
    float* f128  = ws + OFF_F128;
    float* big   = ws + OFF_BIG;
    float* h     = ws + OFF_H;
    float* score = ws + OFF_SCORE;
    float* fm    = ws + OFF_FMAX;
    float* f512  = ws + OFF_F512;
    float* stats = ws + OFF_STATS;

    auto cdiv = [](size_t a, size_t b) { return (int)((a + b - 1) / b); };

    auto bn_pass = [&](float* y, int rows, int C, const float* g, const float* bb,
                       int relu) {
        k_zero<<<cdiv(2 * C, 256), 256, 0, stream>>>(stats, 2 * C);
        size_t tot = (size_t)rows * C;
        int sb = cdiv(tot, 256); if (sb > 4096) sb = 4096;
        k_stats<<<sb, 256, 0, stream>>>(y, rows, C, stats);
        k_bn_apply<<<cdiv(tot, 256), 256, 0, stream>>>(y, rows, C, stats, g, bb, relu);
    };

    auto gemm = [&](const float* A, const float* W, float* O, int R, int Cin,
                    int Cout, int wT) {
        int tiles = (R / 16) * (Cout / 32);
        int blocks = cdiv(tiles, 4);
        switch (Cin) {
        case 64:   k_gemm_wmma<64>  <<<blocks, 128, 0, stream>>>(A, W, O, R, Cout, wT); break;
        case 128:  k_gemm_wmma<128> <<<blocks, 128, 0, stream>>>(A, W, O, R, Cout, wT); break;
        default:   k_gemm_wmma<1024><<<blocks, 128, 0, stream>>>(A, W, O, R, Cout, wT); break;
        }
    };

    auto scorenet = [&](int s) {
        k_linmap<<<cdiv((size_t)BNK * 16, 256), 256, 0, stream>>>(xyz, sn_w1[s], h,
                                                                  BNK, 6, 16);
        bn_pass(h, BNK, 16, sn_g[s], sn_b[s], 1);
        k_score2<<<cdiv(BNK, 256), 256, 0, stream>>>(h, sn_w2[s], sn_b2[s], score, BNK);
    };

    // geometry
    k_build_pts<<<cdiv(BN, 256), 256, 0, stream>>>(x, pts, sq);
    k_knn<<<BN, KNN_T, 0, stream>>>(pts, sq, idx);
    k_xyz<<<cdiv(BNK, 256), 256, 0, stream>>>(pts, idx, xyz);

    // conv1 + bn1 + relu
    k_linmap<<<cdiv((size_t)BN * 64, 256), 256, 0, stream>>>(pts, conv1_w, f64, BN, 3, 64);
    bn_pass(f64, BN, 64, bn1_g, bn1_b, 1);

    // stage 2: f@mat2, scorenet2, assemble, bn2
    gemm(f64, mat2, big, BN, 64, 512, 0);
    scorenet(0);
    k_assemble<<<BN, 64, 0, stream>>>(score, big, idx, f64, 64);
    bn_pass(f64, BN, 64, bn2_g, bn2_b, 1);

    // stage 3
    gemm(f64, mat3, big, BN, 64, 512, 0);
    scorenet(1);
    k_assemble<<<BN, 64, 0, stream>>>(score, big, idx, f64, 64);
    bn_pass(f64, BN, 64, bn3_g, bn3_b, 1);

    // stage 4 (128-wide)
    gemm(f64, mat4, big, BN, 64, 1024, 0);
    scorenet(2);
    k_assemble<<<BN, 128, 0, stream>>>(score, big, idx, f128, 128);
    bn_pass(f128, BN, 128, bn4_g, bn4_b, 1);

    // conv5 (+bn5+relu), max over N
    gemm(f128, conv5_w, big, BN, 128, 1024, 1);
    bn_pass(big, BN, 1024, bn5_g, bn5_b, 1);

    // max-pool into zero-padded 16-row buffer so lin1 can run unpredicated (R=16)
    k_zero<<<cdiv(16 * 1024, 256), 256, 0, stream>>>(fm, 16 * 1024);
    k_colmax<<<cdiv(B_SZ * 1024, 256), 256, 0, stream>>>(big, fm);

    // lin1 (+bn6+relu), lin2
    gemm(fm, lin1_w, f512, 16, 1024, 512, 1);
    bn_pass(f512, B_SZ, 512, bn6_g, bn6_b, 1);
    k_lin2<<<cdiv(B_SZ * 40, 256), 256, 0, stream>>>(f512, lin2_w, lin2_b,
                                                     (float*)d_out);
}